// EdgeToNodeLayer_82162724372841
// MI455X (gfx1250) — compile-verified
//
#include <hip/hip_runtime.h>
#include <hip/hip_bf16.h>

// ---------------------------------------------------------------------------
// EdgeToNode GNN layer for MI455X (gfx1250, wave32, WMMA).
//   agg      = scatter_mean(edge_features by edge_index[1])      [50000 x 64]
//   combined = concat(agg, node_features)                        [50000 x 320]
//   h        = relu(combined @ W1 + b1)                          [50000 x 512]
//   out      = h @ W2 + b2                                       [50000 x 256]
// fp32 throughout -> V_WMMA_F32_16X16X4_F32 (full-precision matrix core).
//
// v2: ROWS=32 (2 M-tiles/block) to double WMMA work per B-fragment load and
// halve L2 weight traffic; float4-vectorized scatter.
// ---------------------------------------------------------------------------

#define N_NODES  50000
#define N_EDGES  800000
#define NODE_DIM 256
#define EDGE_DIM 64
#define IN_DIM   320   // EDGE_DIM + NODE_DIM
#define HIDDEN   512
#define OUT_DIM  256

#define ROWS     32    // node rows per block (2 WMMA M-tiles)
#define CSTRIDE  324   // 320 padded: 324 % 64 == 4 -> conflict-free A-frag reads
#define HSTRIDE  516   // 512 padded: 516 % 64 == 4

typedef __attribute__((ext_vector_type(2))) float v2f;
typedef __attribute__((ext_vector_type(8))) float v8f;

// ---------------------------------------------------------------------------
// Kernel 1: zero agg + counts in workspace.
// ---------------------------------------------------------------------------
__global__ void gnn_zero_ws(float* __restrict__ ws, int n) {
    int i = blockIdx.x * blockDim.x + threadIdx.x;
    int stride = gridDim.x * blockDim.x;
    for (; i < n; i += stride) ws[i] = 0.0f;
}

// ---------------------------------------------------------------------------
// Kernel 2: scatter edge features into per-node sums + counts.
// One thread per (edge, 4-feature chunk): b128 coalesced read + 4 f32 atomics.
// agg buffer (12.8MB) is L2-resident so atomics resolve at L2 atomic units.
// ---------------------------------------------------------------------------
__global__ void gnn_scatter(const int*   __restrict__ tgt,   // edge_index row 1
                            const float* __restrict__ ef,    // [E, 64]
                            float*       __restrict__ agg,   // [N, 64]
                            float*       __restrict__ cnt) { // [N]
    long long idx = (long long)blockIdx.x * blockDim.x + threadIdx.x;
    if (idx >= (long long)N_EDGES * (EDGE_DIM / 4)) return;
    int e = (int)(idx >> 4);          // 16 chunks of 4 per edge
    int q = (int)(idx & 15);
    int f = q * 4;
    int t = tgt[e];
    const float4 v = ((const float4*)ef)[idx];
    float* dst = &agg[t * EDGE_DIM + f];
    atomicAdd(dst + 0, v.x);
    atomicAdd(dst + 1, v.y);
    atomicAdd(dst + 2, v.z);
    atomicAdd(dst + 3, v.w);
    if (q == 0) atomicAdd(&cnt[t], 1.0f);
}

// ---------------------------------------------------------------------------
// Kernel 3: fused 2-layer MLP on 32-row node tiles using f32 WMMA.
// Block: 256 threads = 8 wave32s. LDS: combined tile (41.5KB) + h tile (66KB).
// Each wave: GEMM1 = 2 M-tiles x 4 N-tiles, GEMM2 = 2 M-tiles x 2 N-tiles.
// ---------------------------------------------------------------------------
__launch_bounds__(256)
__global__ void gnn_fused_mlp(const float* __restrict__ nf,   // [N, 256]
                              const float* __restrict__ agg,  // [N, 64] sums
                              const float* __restrict__ cnt,  // [N]
                              const float* __restrict__ W1,   // [320, 512]
                              const float* __restrict__ b1,   // [512]
                              const float* __restrict__ W2,   // [512, 256]
                              const float* __restrict__ b2,   // [256]
                              float*       __restrict__ out) {// [N, 256]
    __shared__ float sC[ROWS * CSTRIDE];  // combined tile
    __shared__ float sH[ROWS * HSTRIDE];  // hidden tile

    const int tid   = threadIdx.x;
    const int node0 = blockIdx.x * ROWS;

    // ---- Stage 0: build combined[32 x 320] in LDS (mean-scale fused) ------
    for (int i = tid; i < ROWS * IN_DIM; i += 256) {
        int r = i / IN_DIM;
        int k = i - r * IN_DIM;
        int node = node0 + r;
        if (node >= N_NODES) node = N_NODES - 1;   // tail clamp (rows unused)
        float v;
        if (k < EDGE_DIM) {
            float c = cnt[node];
            c = (c < 1.0f) ? 1.0f : c;
            v = agg[node * EDGE_DIM + k] / c;
        } else {
            v = nf[node * NODE_DIM + (k - EDGE_DIM)];
        }
        sC[r * CSTRIDE + k] = v;
    }
    __syncthreads();

    const int wave = tid >> 5;
    const int lane = tid & 31;
    const int half = lane >> 4;   // 0: lanes 0-15, 1: lanes 16-31
    const int m    = lane & 15;

    // ---- GEMM1: h = relu(combined @ W1 + b1) ------------------------------
    {
        v8f acc[2][4] = {};
        const int nbase = wave * 64;                 // 4 N-tiles of 16
        for (int k0 = 0; k0 < IN_DIM; k0 += 4) {
            // A frag 16x4 f32: lane<16 -> (M=lane, K=k0+{0,1}),
            //                  lane>=16 -> (M=lane-16, K=k0+{2,3})
            v2f a0 = *(const v2f*)&sC[ m       * CSTRIDE + k0 + 2 * half];
            v2f a1 = *(const v2f*)&sC[(m + 16) * CSTRIDE + k0 + 2 * half];
#pragma unroll
            for (int j = 0; j < 4; ++j) {
                int n = nbase + j * 16 + m;
                v2f b;
                b.x = W1[(k0 + 2 * half)     * HIDDEN + n];
                b.y = W1[(k0 + 2 * half + 1) * HIDDEN + n];
                acc[0][j] = __builtin_amdgcn_wmma_f32_16x16x4_f32(
                    false, a0, false, b, (short)0, acc[0][j], false, false);
                acc[1][j] = __builtin_amdgcn_wmma_f32_16x16x4_f32(
                    false, a1, false, b, (short)0, acc[1][j], false, false);
            }
        }
        // Epilogue: C/D layout -> VGPR v holds M = v + 8*half, N = m.
#pragma unroll
        for (int t = 0; t < 2; ++t) {
#pragma unroll
            for (int j = 0; j < 4; ++j) {
                int n = nbase + j * 16 + m;
                float bias = b1[n];
#pragma unroll
                for (int v = 0; v < 8; ++v) {
                    int M = v + 8 * half + 16 * t;
                    float val = acc[t][j][v] + bias;
                    sH[M * HSTRIDE + n] = (val > 0.0f) ? val : 0.0f;
                }
            }
        }
    }
    __syncthreads();

    // ---- GEMM2: out = h @ W2 + b2 ----------------------------------------
    {
        v8f acc[2][2] = {};
        const int nbase = wave * 32;                 // 2 N-tiles of 16
        for (int k0 = 0; k0 < HIDDEN; k0 += 4) {
            v2f a0 = *(const v2f*)&sH[ m       * HSTRIDE + k0 + 2 * half];
            v2f a1 = *(const v2f*)&sH[(m + 16) * HSTRIDE + k0 + 2 * half];
#pragma unroll
            for (int j = 0; j < 2; ++j) {
                int n = nbase + j * 16 + m;
                v2f b;
                b.x = W2[(k0 + 2 * half)     * OUT_DIM + n];
                b.y = W2[(k0 + 2 * half + 1) * OUT_DIM + n];
                acc[0][j] = __builtin_amdgcn_wmma_f32_16x16x4_f32(
                    false, a0, false, b, (short)0, acc[0][j], false, false);
                acc[1][j] = __builtin_amdgcn_wmma_f32_16x16x4_f32(
                    false, a1, false, b, (short)0, acc[1][j], false, false);
            }
        }
#pragma unroll
        for (int t = 0; t < 2; ++t) {
#pragma unroll
            for (int j = 0; j < 2; ++j) {
                int n = nbase + j * 16 + m;
                float bias = b2[n];
#pragma unroll
                for (int v = 0; v < 8; ++v) {
                    int M = v + 8 * half + 16 * t;
                    int node = node0 + M;
                    if (node < N_NODES)
                        out[node * OUT_DIM + n] = acc[t][j][v] + bias;
                }
            }
        }
    }
}

// ---------------------------------------------------------------------------
// Launcher. Inputs (setup_inputs order):
//   0 node_features f32 [50000,256]   1 edge_index i32 [2,800000]
//   2 edge_features f32 [800000,64]   3 W1 f32 [320,512]   4 b1 f32 [512]
//   5 W2 f32 [512,256]                6 b2 f32 [256]
// d_ws layout: agg [50000*64] f32, then cnt [50000] f32  (~13 MB).
// ---------------------------------------------------------------------------
extern "C" void kernel_launch(void* const* d_in, const int* in_sizes, int n_in,
                              void* d_out, int out_size, void* d_ws, size_t ws_size,
                              hipStream_t stream) {
    const float* nf = (const float*)d_in[0];
    const int*   ei = (const int*)d_in[1];
    const float* ef = (const float*)d_in[2];
    const float* W1 = (const float*)d_in[3];
    const float* b1 = (const float*)d_in[4];
    const float* W2 = (const float*)d_in[5];
    const float* b2 = (const float*)d_in[6];
    float* out = (float*)d_out;

    float* agg = (float*)d_ws;
    float* cnt = agg + (size_t)N_NODES * EDGE_DIM;

    // 1) zero accumulators
    int zn = N_NODES * EDGE_DIM + N_NODES;
    gnn_zero_ws<<<2048, 256, 0, stream>>>(agg, zn);

    // 2) scatter-sum edges (edge_index row 1 = targets), float4 per thread
    const int* tgt = ei + N_EDGES;
    long long total = (long long)N_EDGES * (EDGE_DIM / 4);
    int sblocks = (int)((total + 255) / 256);
    gnn_scatter<<<sblocks, 256, 0, stream>>>(tgt, ef, agg, cnt);

    // 3) fused MLP over 32-row node tiles (ceil(50000/32) = 1563, guarded tail)
    int gblocks = (N_NODES + ROWS - 1) / ROWS;
    gnn_fused_mlp<<<gblocks, 256, 0, stream>>>(
        nf, agg, cnt, W1, b1, W2, b2, out);
}